// CenterAttentionLoss_3753801417247
// MI455X (gfx1250) — compile-verified
//
#include <hip/hip_runtime.h>

// CDNA5 / gfx1250, wave32. Pairwise-distance-as-GEMM via V_WMMA_F32_16X16X4_F32.
// heat[p] = exp(-min_n d2(p,n)/8); d2 decomposed as rank-3 GEMM:
//   D[M][N] = aug_N - 2(px_M*cx_N + py_M*cy_N),  d2 = D_min + px^2+py^2
// A row = (-2px, -2py, 1, 0); B col = (cx, cy, cx^2+cy^2, 0) or (0,0,BIG,0).

typedef float v2f __attribute__((ext_vector_type(2)));
typedef float v8f __attribute__((ext_vector_type(8)));

#define NB 4
#define NP 4096      // H*W = 64*64
#define NTILE 256    // NP / 16 column tiles
#define BIGV 1.0e9f  // sentinel for unmasked centers (real d^2 - p^2 is within +-8000)

__global__ void zero_out_kernel(float* out) { out[0] = 0.0f; }

__global__ __launch_bounds__(1024)
void center_attention_loss_kernel(const float* __restrict__ att,   // [B,1,H,W]
                                  const int*   __restrict__ cls,   // [B,H,W]
                                  const float* __restrict__ box,   // [B,H,W,2]
                                  float* __restrict__ out)
{
    // Packed B-operand staging: slot 2n+0 = (cx,cy) for K0/K1 (lanes 0-15),
    //                           slot 2n+1 = (aug,0) for K2/K3 (lanes 16-31).
    // One unconditional ds_load_b64 per lane per tile, conflict-free banks.
    __shared__ float2 sB[NP * 2];      // 64 KB

    const int b   = blockIdx.x >> 3;   // 8 blocks per batch
    const int g   = blockIdx.x & 7;
    const int tid = threadIdx.x;

    // --- Stage all 4096 centers of batch b into LDS (uniform loop, no divergence) ---
    for (int n = tid; n < NP; n += 1024) {
        const float cx = box[((size_t)b * NP + n) * 2 + 0];
        const float cy = box[((size_t)b * NP + n) * 2 + 1];
        const bool  m  = cls[(size_t)b * NP + n] > 0;
        float2 lo, hi;
        lo.x = m ? cx : 0.0f;
        lo.y = m ? cy : 0.0f;
        hi.x = m ? __builtin_fmaf(cx, cx, cy * cy) : BIGV;
        hi.y = 0.0f;
        sB[(n << 1) + 0] = lo;
        sB[(n << 1) + 1] = hi;
    }
    __syncthreads();

    // --- Per-wave strip of 16 consecutive pixels ---
    const int wv   = tid >> 5;        // wave id in block [0,32)
    const int lane = tid & 31;
    const int half = lane >> 4;       // 0: lanes 0-15 (K0,K1) ; 1: lanes 16-31 (K2,K3)
    const int sub  = lane & 15;       // row (A) / column (B,D) index

    const int   s   = (g << 5) | wv;  // strip id in [0,256)
    const int   p0  = s << 4;         // first pixel of strip (16-aligned, same image row)
    const float fy  = (float)(p0 >> 6);
    const float fx0 = (float)(p0 & 63);

    // A matrix (16x4 f32): row M=sub is (-2*px, -2*py, 1, 0)
    // lanes 0-15: VGPR0=K0, VGPR1=K1 ; lanes 16-31: VGPR0=K2, VGPR1=K3
    const float rowpx = fx0 + (float)sub;
    v2f Av;
    Av.x = half ? 1.0f : (-2.0f * rowpx);
    Av.y = half ? 0.0f : (-2.0f * fy);

    const v8f czero = {0.f,0.f,0.f,0.f,0.f,0.f,0.f,0.f};
    float rmin[8];
#pragma unroll
    for (int i = 0; i < 8; ++i) rmin[i] = 4.0f * BIGV;

    // Per-lane constant slot: 2*sub + half (all 32 lanes distinct)
    const float2* __restrict__ pB = &sB[(sub << 1) | half];

    // --- 256 WMMA tiles with running per-lane min ---
#pragma unroll 8
    for (int t = 0; t < NTILE; ++t) {
        const float2 bp = pB[t << 5];   // immediate DS offset t*256 bytes
        v2f Bv;
        Bv.x = bp.x;                    // K0=cx (lo) / K2=aug (hi)
        Bv.y = bp.y;                    // K1=cy (lo) / K3=0   (hi)
        // 8 args: (neg_a, A, neg_b, B, c_mod, C, reuse_a, reuse_b)
        const v8f d = __builtin_amdgcn_wmma_f32_16x16x4_f32(
            false, Av, false, Bv, (short)0, czero, false, false);
#pragma unroll
        for (int i = 0; i < 8; ++i) rmin[i] = fminf(rmin[i], d[i]);
    }

    // --- min across the 16 N-columns (xor masks < 16 stay within each half-wave) ---
#pragma unroll
    for (int i = 0; i < 8; ++i) {
        float v = rmin[i];
        v = fminf(v, __shfl_xor(v, 1, 32));
        v = fminf(v, __shfl_xor(v, 2, 32));
        v = fminf(v, __shfl_xor(v, 4, 32));
        v = fminf(v, __shfl_xor(v, 8, 32));
        rmin[i] = v;
    }

    // D layout: VGPR i holds row i (lanes 0-15) and row i+8 (lanes 16-31).
    // After reduction lane 0 owns rows 0-7, lane 16 owns rows 8-15.
    if (sub == 0) {
        float acc = 0.0f;
#pragma unroll
        for (int i = 0; i < 8; ++i) {
            const int   p  = p0 + i + (half << 3);
            const float px = (float)(p & 63);
            const float py = (float)(p >> 6);
            const float d2 = rmin[i] + __builtin_fmaf(px, px, py * py);
            // heat = exp(-d2 / (2*sigma^2)) = exp(-0.125*d2); 0 if no masked center
            const float heat = (rmin[i] < 0.5f * BIGV) ? __expf(-0.125f * d2) : 0.0f;
            const float a    = att[(size_t)b * NP + p];
            const float sig  = 1.0f / (1.0f + __expf(-a));
            const float df   = sig - heat;
            acc = __builtin_fmaf(df, df, acc);
        }
        // mean over B*H*W (=16384) then * WEIGHT (=0.05)
        atomicAdd(out, acc * (0.05f / 16384.0f));
    }
}

extern "C" void kernel_launch(void* const* d_in, const int* in_sizes, int n_in,
                              void* d_out, int out_size, void* d_ws, size_t ws_size,
                              hipStream_t stream) {
    (void)in_sizes; (void)n_in; (void)out_size; (void)d_ws; (void)ws_size;
    const float* att = (const float*)d_in[0];   // [4,1,64,64] f32
    const int*   cls = (const int*)  d_in[1];   // [4,64,64]   i32
    const float* box = (const float*)d_in[2];   // [4,64,64,2] f32
    float* out = (float*)d_out;

    zero_out_kernel<<<1, 1, 0, stream>>>(out);
    // 4 batches * 8 block-groups; 1024 threads = 32 wave32 strips per block
    center_attention_loss_kernel<<<dim3(NB * 8), dim3(1024), 0, stream>>>(att, cls, box, out);
}